// BertSelfAttention_61194694034283
// MI455X (gfx1250) — compile-verified
//
#include <hip/hip_runtime.h>
#include <math.h>

#define BATCH 8
#define SEQ   4096
#define SEQP1 4097
#define HID   1024
#define NHEAD 16
#define HDIM  64

typedef float v2f __attribute__((ext_vector_type(2)));
typedef float v8f __attribute__((ext_vector_type(8)));

// ---------------- Kernel A: q = cell @ Wq^T + bq  -> (B, H) ----------------
__global__ void qproj_kernel(const float* __restrict__ cell,
                             const float* __restrict__ Wq,
                             const float* __restrict__ bq,
                             float* __restrict__ q) {
  int g = blockIdx.x * blockDim.x + threadIdx.x;   // 0 .. B*H-1
  int b = g >> 10;
  int j = g & 1023;
  const float* wrow = Wq + (size_t)j * HID;
  const float* crow = cell + (size_t)b * HID;
  float acc = 0.f;
#pragma unroll 8
  for (int i = 0; i < HID; ++i) acc = fmaf(wrow[i], crow[i], acc);
  q[g] = acc + bq[j];
}

// --- Kernel A2: w~[b,:,h] = Wk_h^T q[b,h]  (B, HID, NHEAD); c[b,h] = q.bk ---
__global__ void wtilde_kernel(const float* __restrict__ q,
                              const float* __restrict__ Wk,
                              const float* __restrict__ bk,
                              float* __restrict__ wt,   // [B][HID][NHEAD]
                              float* __restrict__ cvec) // [B][NHEAD]
{
  int b = blockIdx.x >> 4;
  int h = blockIdx.x & 15;
  __shared__ float qs[HDIM];
  int tid = threadIdx.x;                 // 256 threads
  if (tid < HDIM) qs[tid] = q[b * HID + h * HDIM + tid];
  __syncthreads();
  for (int j = tid; j < HID; j += 256) {
    float acc = 0.f;
#pragma unroll 8
    for (int d = 0; d < HDIM; ++d)
      acc = fmaf(qs[d], Wk[(size_t)(h * HDIM + d) * HID + j], acc);
    wt[((size_t)b * HID + j) * NHEAD + h] = acc;
  }
  if (tid == 0) {
    float acc = 0.f;
    for (int d = 0; d < HDIM; ++d) acc = fmaf(qs[d], bk[h * HDIM + d], acc);
    cvec[b * NHEAD + h] = acc;
  }
}

// ---- Kernel B: scores[b,h,t] = (X[b,t,:].w~[b,:,h] + c[b,h]) / 8  (WMMA) ----
// Per wave: one 16-row t-tile, all 16 heads. A = X tile (16x4), B = w~ (4x16).
__global__ void __launch_bounds__(128)
scores_kernel(const float* __restrict__ hs,
              const float* __restrict__ wt,
              const float* __restrict__ cvec,
              float* __restrict__ scores)  // [B][NHEAD][SEQ]
{
  int wave = threadIdx.x >> 5;
  int lane = threadIdx.x & 31;
  int tile = blockIdx.x * 4 + wave;     // 0 .. B*(SEQ/16)-1
  int b  = tile >> 8;                   // 256 tiles per batch
  int t0 = (tile & 255) << 4;
  int m  = lane & 15;                   // M row (t) / N col (head)
  int kh = lane >> 4;                   // K half select
  const float* arow = hs + ((size_t)b * SEQP1 + (t0 + m)) * HID + 2 * kh;
  const float* brow = wt + (size_t)b * HID * NHEAD;
  v8f acc = {};
#pragma unroll 4
  for (int k = 0; k < HID; k += 4) {
    v2f a = *(const v2f*)(arow + k);                 // A: K=2kh, 2kh+1 (contiguous)
    v2f bb;
    bb.x = brow[(k + 2 * kh) * NHEAD + m];           // B row K=2kh, col m
    bb.y = brow[(k + 2 * kh + 1) * NHEAD + m];       // B row K=2kh+1, col m
    acc = __builtin_amdgcn_wmma_f32_16x16x4_f32(false, a, false, bb,
                                                (short)0, acc, false, false);
  }
  float c = cvec[b * NHEAD + m];
  float* srow = scores + ((size_t)b * NHEAD + m) * SEQ + t0 + 8 * kh;
#pragma unroll
  for (int v = 0; v < 8; ++v)
    srow[v] = (acc[v] + c) * 0.125f;   // 1/sqrt(64)
}

// ---------------- Kernel C: masked softmax over t, in place ----------------
__global__ void softmax_kernel(float* __restrict__ s, const int* __restrict__ mask) {
  __shared__ float red[256];
  int row = blockIdx.x;                 // b*NHEAD + h
  int b = row >> 4;
  float* sr = s + (size_t)row * SEQ;
  const int* mr = mask + (size_t)b * SEQ;
  int tid = threadIdx.x;
  float vals[16];
  float mx = -INFINITY;
#pragma unroll
  for (int i = 0; i < 16; ++i) {
    int t = tid + (i << 8);
    float x = sr[t];
    x = (mr[t] == 0) ? -INFINITY : x;
    vals[i] = x;
    mx = fmaxf(mx, x);
  }
  red[tid] = mx;
  __syncthreads();
  for (int o = 128; o > 0; o >>= 1) {
    if (tid < o) red[tid] = fmaxf(red[tid], red[tid + o]);
    __syncthreads();
  }
  mx = red[0];
  __syncthreads();
  float sum = 0.f;
#pragma unroll
  for (int i = 0; i < 16; ++i) { vals[i] = __expf(vals[i] - mx); sum += vals[i]; }
  red[tid] = sum;
  __syncthreads();
  for (int o = 128; o > 0; o >>= 1) {
    if (tid < o) red[tid] += red[tid + o];
    __syncthreads();
  }
  float inv = 1.0f / red[0];
#pragma unroll
  for (int i = 0; i < 16; ++i) sr[tid + (i << 8)] = vals[i] * inv;
}

// ------- Kernel D: Y[b,h,:] = sum_t P[b,h,t] * X[b,t,:]   (WMMA, K = t) -----
// Block = (b, 16-col tile); 4 waves split the t range, LDS-reduce.
__global__ void __launch_bounds__(128)
wsum_kernel(const float* __restrict__ hs,
            const float* __restrict__ probs,  // [B][NHEAD][SEQ]
            float* __restrict__ y)            // [B][NHEAD][HID]
{
  __shared__ float red[4 * 256];
  int wave = threadIdx.x >> 5;
  int lane = threadIdx.x & 31;
  int b  = blockIdx.x >> 6;             // 64 column tiles per batch
  int c0 = (blockIdx.x & 63) << 4;
  int m  = lane & 15;
  int kh = lane >> 4;
  int tbeg = wave * (SEQ / 4);
  const float* prow = probs + ((size_t)b * NHEAD + m) * SEQ + 2 * kh; // A: P (16h x 4t)
  const float* xcol = hs + (size_t)b * SEQP1 * HID + c0 + m;          // B: X (4t x 16c)
  v8f acc = {};
#pragma unroll 4
  for (int t = tbeg; t < tbeg + SEQ / 4; t += 4) {
    v2f a = *(const v2f*)(prow + t);
    v2f bb;
    bb.x = xcol[(size_t)(t + 2 * kh) * HID];
    bb.y = xcol[(size_t)(t + 2 * kh + 1) * HID];
    acc = __builtin_amdgcn_wmma_f32_16x16x4_f32(false, a, false, bb,
                                                (short)0, acc, false, false);
  }
#pragma unroll
  for (int v = 0; v < 8; ++v) red[wave * 256 + v * 32 + lane] = acc[v];
  __syncthreads();
  if (wave == 0) {
#pragma unroll
    for (int v = 0; v < 8; ++v) {
      float s = red[v * 32 + lane] + red[256 + v * 32 + lane] +
                red[512 + v * 32 + lane] + red[768 + v * 32 + lane];
      int h = v + 8 * kh;                                  // D: M=head, N=col
      y[((size_t)b * NHEAD + h) * HID + c0 + m] = s;
    }
  }
}

// ------------- Kernel E: out[b,r] = Wv[r,:].Y[b, r/64, :] + bv[r] -----------
__global__ void outproj_kernel(const float* __restrict__ Wv,
                               const float* __restrict__ bv,
                               const float* __restrict__ y,
                               float* __restrict__ out) {
  int g = blockIdx.x * blockDim.x + threadIdx.x;  // 0 .. B*HID-1
  int b = g >> 10;
  int r = g & 1023;
  int h = r >> 6;
  const float* wrow = Wv + (size_t)r * HID;
  const float* yrow = y + ((size_t)b * NHEAD + h) * HID;
  float acc = 0.f;
#pragma unroll 8
  for (int i = 0; i < HID; ++i) acc = fmaf(wrow[i], yrow[i], acc);
  out[g] = acc + bv[r];
}

extern "C" void kernel_launch(void* const* d_in, const int* in_sizes, int n_in,
                              void* d_out, int out_size, void* d_ws, size_t ws_size,
                              hipStream_t stream) {
  (void)in_sizes; (void)n_in; (void)out_size; (void)ws_size;
  const float* hs   = (const float*)d_in[0];   // (B, T+1, H)
  const float* cell = (const float*)d_in[1];   // (B, H)
  const float* Wq   = (const float*)d_in[2];
  const float* bq   = (const float*)d_in[3];
  const float* Wk   = (const float*)d_in[4];
  const float* bk   = (const float*)d_in[5];
  const float* Wv   = (const float*)d_in[6];
  const float* bv   = (const float*)d_in[7];
  const int*   am   = (const int*)d_in[8];     // (B, T)
  float* out = (float*)d_out;                  // (B, H)

  float* ws   = (float*)d_ws;
  float* q    = ws;                                   // B*H          =   8K f
  float* wt   = q + BATCH * HID;                      // B*H*NH       = 128K f
  float* cvec = wt + (size_t)BATCH * HID * NHEAD;     // B*NH
  float* sc   = cvec + BATCH * NHEAD;                 // B*NH*SEQ     = 512K f
  float* y    = sc + (size_t)BATCH * NHEAD * SEQ;     // B*NH*H       = 128K f
  // total ~3.1 MB of workspace

  qproj_kernel  <<<BATCH * HID / 256, 256, 0, stream>>>(cell, Wq, bq, q);
  wtilde_kernel <<<BATCH * NHEAD,      256, 0, stream>>>(q, Wk, bk, wt, cvec);
  scores_kernel <<<BATCH * (SEQ / 16) / 4, 128, 0, stream>>>(hs, wt, cvec, sc);
  softmax_kernel<<<BATCH * NHEAD,      256, 0, stream>>>(sc, am);
  wsum_kernel   <<<BATCH * (HID / 16), 128, 0, stream>>>(hs, sc, y);
  outproj_kernel<<<BATCH * HID / 256,  256, 0, stream>>>(Wv, bv, y, out);
}